// SolitonStability_76441827934260
// MI455X (gfx1250) — compile-verified
//
#include <hip/hip_runtime.h>
#include <hip/hip_bf16.h>

// CDNA5 / gfx1250: wave32. f32 WMMA used for cross-lane reductions.
typedef __attribute__((ext_vector_type(2))) float v2f;
typedef __attribute__((ext_vector_type(4))) float f4;
typedef __attribute__((ext_vector_type(8))) float v8f;

#define ETA_C   0.5f
#define KAPPA_C 0.1f

// Full 32-lane sum using V_WMMA_F32_16X16X4_F32 (f32 matrix pipe, RNE).
// A layout (ISA 7.12.2, 32-bit A 16x4): lane l<16: v0=A[l][0], v1=A[l][1];
// lane l>=16: v0=A[l-16][2], v1=A[l-16][3].
// We place the per-lane partial in v0 and zero in v1; B = all-ones, so
// D[m][n] = p_m + p_{m+16} for every n. D layout: vgpr r holds row r
// (lanes 0-15) and row r+8 (lanes 16-31). Summing the 8 D vgprs gives
// lanes <16: sum_{m=0..7}(p_m+p_{m+16}), lanes >=16: sum_{m=8..15}(...).
// One xor-16 swap + add yields the full 32-lane total in every lane.
// EXEC must be all ones at the call site (uniform control flow only).
__device__ __forceinline__ float wave_reduce32(float p) {
    v2f a;    a.x = p;    a.y = 0.0f;
    v2f ones; ones.x = 1.0f; ones.y = 1.0f;
    v8f c = {};
    v8f d = __builtin_amdgcn_wmma_f32_16x16x4_f32(
        /*neg_a=*/false, a, /*neg_b=*/false, ones,
        /*c_mod=*/(short)0, c, /*reuse_a=*/false, /*reuse_b=*/false);
    float t = ((d[0] + d[1]) + (d[2] + d[3])) + ((d[4] + d[5]) + (d[6] + d[7]));
    t += __shfl_xor(t, 16);
    return t;
}

__global__ void __launch_bounds__(512)
soliton_stability_kernel(const float* __restrict__ residue,
                         const float* __restrict__ constraint,
                         const float* __restrict__ pert,
                         float* __restrict__ out,
                         int B, int D, int S)
{
    const int b    = blockIdx.x;
    const int tid  = threadIdx.x;
    const int lane = tid & 31;
    const int wid  = tid >> 5;
    const int nthr = blockDim.x;

    // ---- dominant stream: sum of squares over perturbations[b] (S*D floats) ----
    // 420 MB total across the grid > 192 MB L2 -> non-temporal loads.
    const f4* p4 = (const f4*)(pert + (size_t)b * (size_t)S * (size_t)D);
    const int n4 = (S * D) >> 2;
    float ax = 0.f, ay = 0.f, az = 0.f, aw = 0.f;
    for (int i = tid; i < n4; i += nthr) {
        f4 v = __builtin_nontemporal_load(p4 + i);
        ax = fmaf(v.x, v.x, ax);
        ay = fmaf(v.y, v.y, ay);
        az = fmaf(v.z, v.z, az);
        aw = fmaf(v.w, v.w, aw);
    }
    float pp = (ax + ay) + (az + aw);

    // ---- small per-row streams: residue & constraint (D floats each) ----
    const f4* r4 = (const f4*)(residue    + (size_t)b * (size_t)D);
    const f4* c4 = (const f4*)(constraint + (size_t)b * (size_t)D);
    const int d4 = D >> 2;
    float rr = 0.f, cc = 0.f;
    for (int i = tid; i < d4; i += nthr) {
        f4 v = r4[i];
        rr += v.x * v.x + v.y * v.y + v.z * v.z + v.w * v.w;
        f4 u = c4[i];
        cc += u.x * u.x + u.y * u.y + u.z * u.z + u.w * u.w;
    }

    // ---- wave-level reduction on the f32 matrix pipe (EXEC all ones here) ----
    pp = wave_reduce32(pp);
    rr = wave_reduce32(rr);
    cc = wave_reduce32(cc);

    __shared__ float sP[16], sR[16], sC[16];
    if (lane == 0) { sP[wid] = pp; sR[wid] = rr; sC[wid] = cc; }
    __syncthreads();

    if (wid == 0) {
        const int nw = nthr >> 5;  // 16 waves
        float vp = (lane < nw) ? sP[lane] : 0.0f;  // select, no EXEC change
        float vr = (lane < nw) ? sR[lane] : 0.0f;
        float vc = (lane < nw) ? sC[lane] : 0.0f;
        vp = wave_reduce32(vp);
        vr = wave_reduce32(vr);
        vc = wave_reduce32(vc);
        if (lane == 0) {
            float dispersion     = vp / (float)S;          // mean over samples
            float energy         = vr;
            float scale          = sqrtf(vc);              // ||constraint||_2
            float energy_density = energy / (scale + 1e-8f);
            float localization   = (energy > 0.0f) ? (ETA_C / (energy_density + 1e-8f))
                                                   : scale;
            float ratio          = dispersion / (localization + 1e-8f);
            out[b]             = (ratio < KAPPA_C) ? 1.0f : 0.0f;  // is_soliton
            out[B + b]         = dispersion;
            out[2 * B + b]     = localization;
            out[3 * B + b]     = ratio;
        }
    }
}

extern "C" void kernel_launch(void* const* d_in, const int* in_sizes, int n_in,
                              void* d_out, int out_size, void* d_ws, size_t ws_size,
                              hipStream_t stream) {
    (void)n_in; (void)out_size; (void)d_ws; (void)ws_size;
    const float* residue    = (const float*)d_in[0];  // [B, D] f32
    const float* constraint = (const float*)d_in[1];  // [B, D] f32
    const float* pert       = (const float*)d_in[2];  // [B, S, D] f32
    float* out = (float*)d_out;                       // [4*B] f32 (bool as 0/1)

    const int D = 1024;
    const int B = in_sizes[0] / D;          // 1024
    const int S = in_sizes[2] / in_sizes[0]; // 100

    soliton_stability_kernel<<<B, 512, 0, stream>>>(residue, constraint, pert,
                                                    out, B, D, S);
}